// EVRInitEmbedding_36799279792448
// MI455X (gfx1250) — compile-verified
//
#include <hip/hip_runtime.h>
#include <hip/hip_bf16.h>

// Problem constants (from reference)
#define BATCH   32
#define NNODE   2000
#define NEDGE   64000
#define EMBED   128
#define INFEAT  6
#define GROUPS  (NNODE / 16)      // 125 node-groups of 16 per batch
#define EPB     8000              // edges per scatter block
#define NBLK_E  (NEDGE / EPB)     // 8 scatter blocks per batch

typedef float v2f __attribute__((ext_vector_type(2)));
typedef float v8f __attribute__((ext_vector_type(8)));

// ---------------------------------------------------------------------------
// Kernel 0: zero the per-node accumulator workspace.
// ws layout: sums[b][n][8] : {sum_xsrc0, sum_xsrc1, sum_ea0, sum_ea1, cnt, pad...}
// ---------------------------------------------------------------------------
__global__ void zero_ws_kernel(float* __restrict__ s, int n) {
    int i = blockIdx.x * blockDim.x + threadIdx.x;
    if (i < n) s[i] = 0.0f;
}

// ---------------------------------------------------------------------------
// Kernel 1: edge scatter with LDS pre-aggregation.
// grid = (NBLK_E, BATCH), block = 256. Each block owns EPB edges of one batch,
// accumulates into a 40 KB LDS histogram (ds_add_f32), then flushes with
// global f32 atomics (at least 4x fewer global atomics than direct scatter).
// ---------------------------------------------------------------------------
__global__ __launch_bounds__(256) void edge_scatter_kernel(
    const float* __restrict__ locs,   // [B][N][2]
    const int*   __restrict__ ei,     // [B][2][E]
    const float* __restrict__ ea,     // [B][E][2]
    float*       __restrict__ sums)   // [B][N][8]
{
    __shared__ float s[NNODE * 5];
    const int tid = threadIdx.x;

    for (int i = tid; i < NNODE * 5; i += 256) s[i] = 0.0f;
    __syncthreads();

    const int b = blockIdx.y;
    const int*   eib = ei   + (size_t)b * 2 * NEDGE;
    const float* lb  = locs + (size_t)b * NNODE * 2;
    const float* eab = ea   + (size_t)b * NEDGE * 2;

    const int base = blockIdx.x * EPB;
    for (int e = base + tid; e < base + EPB; e += 256) {
        const int src = eib[e];
        const int tgt = eib[NEDGE + e];
        const float x0 = lb[src * 2 + 0];
        const float x1 = lb[src * 2 + 1];
        const float a0 = eab[e * 2 + 0];
        const float a1 = eab[e * 2 + 1];
        atomicAdd(&s[tgt * 5 + 0], x0);
        atomicAdd(&s[tgt * 5 + 1], x1);
        atomicAdd(&s[tgt * 5 + 2], a0);
        atomicAdd(&s[tgt * 5 + 3], a1);
        atomicAdd(&s[tgt * 5 + 4], 1.0f);
    }
    __syncthreads();

    float* sb = sums + (size_t)b * NNODE * 8;
    for (int i = tid; i < NNODE * 5; i += 256) {
        const float v = s[i];
        if (v != 0.0f) {
            const int node = i / 5;
            const int j    = i - node * 5;
            atomicAdd(&sb[node * 8 + j], v);
        }
    }
}

// ---------------------------------------------------------------------------
// Kernel 2: WMMA f32 16x16x4 GEMM: out[16 nodes x 128] = meanfeats[16x8] @ Wt[8x128] (+b, masked)
// One wave per (batch, 16-node group); 4 waves / block; 1000 blocks = 4000 waves exactly.
// A layout (32-bit 16x4): lanes 0-15 -> {K0,K1} of row M=lane; lanes 16-31 -> {K2,K3}.
// B layout (4x16):        lanes 0-15 -> {K0,K1} of col N=lane; lanes 16-31 -> {K2,K3}.
// C/D layout:             VGPR r -> M = r + 8*(lane>=16), N = lane&15.
// Branch-free operand setup: index arithmetic + register selects/mask-mul only,
// so EXEC stays all-ones with no saveexec regions between the loads and WMMAs.
// ---------------------------------------------------------------------------
__global__ __launch_bounds__(128) void node_gemm_wmma_kernel(
    const float* __restrict__ locs,   // [B][N][2]
    const float* __restrict__ sums,   // [B][N][8]
    const float* __restrict__ W,      // [128][6] row-major
    const float* __restrict__ bias,   // [128]
    float*       __restrict__ out)    // [B][N][128]
{
    const int wave = blockIdx.x * (blockDim.x >> 5) + (threadIdx.x >> 5);
    const int lane = threadIdx.x & 31;
    const int half = lane >> 4;       // 0: K={0,1}/{4,5}, 1: K={2,3}/{6,7}
    const int lid  = lane & 15;

    const int b = wave / GROUPS;
    const int g = wave - b * GROUPS;

    // ---- A operand: meanfeats row for node n = g*16 + lid ----
    const int n = g * 16 + lid;
    const float* srow = sums + ((size_t)b * NNODE + n) * 8;
    const float cnt = srow[4];
    const float inv = 1.0f / fmaxf(cnt, 1.0f);

    const float lx0 = locs[((size_t)b * NNODE + n) * 2 + 0];
    const float lx1 = locs[((size_t)b * NNODE + n) * 2 + 1];
    const float mx0 = srow[0] * inv;  // mean x_src[0]  (K=2)
    const float mx1 = srow[1] * inv;  // mean x_src[1]  (K=3)
    const float me0 = srow[2] * inv;  // mean ea[0]     (K=4)
    const float me1 = srow[3] * inv;  // mean ea[1]     (K=5)

    const float km = half ? 0.0f : 1.0f;  // zero-pad mask for K=6,7 (lanes 16-31, 2nd WMMA)

    v2f a0, a1;                       // K-halves 0..3 and 4..7 (6,7 zero-padded)
    a0.x = half ? mx0 : lx0;          // register cndmask, no branch
    a0.y = half ? mx1 : lx1;
    a1.x = me0 * km;
    a1.y = me1 * km;

    // counts for the 8 output rows this lane owns (M = half*8 + r):
    // row M's count is held by lane M -- broadcast via lane permute, no reloads.
    float cm[8];
    #pragma unroll
    for (int r = 0; r < 8; ++r)
        cm[r] = __shfl(cnt, half * 8 + r, 32);

    // ---- 8 embed tiles of 16 columns each ----
    #pragma unroll
    for (int t = 0; t < 8; ++t) {
        const int col  = t * 16 + lid;           // embed column, B-matrix N index
        const int kbase = col * 6 + 2 * half;    // Wt[k][col] = W[col][k]
        v2f b0, b1;
        b0.x = W[kbase + 0];                     // unconditional paired load
        b0.y = W[kbase + 1];
        b1.x = W[col * 6 + 4] * km;              // unconditional load + mask
        b1.y = W[col * 6 + 5] * km;

        v8f c = {};
        c = __builtin_amdgcn_wmma_f32_16x16x4_f32(false, a0, false, b0, (short)0, c, false, false);
        c = __builtin_amdgcn_wmma_f32_16x16x4_f32(false, a1, false, b1, (short)0, c, false, false);

        const float bv = bias[col];
        #pragma unroll
        for (int r = 0; r < 8; ++r) {
            const int nm = g * 16 + half * 8 + r;
            const float val = (cm[r] > 0.0f) ? (c[r] + bv) : 0.0f;
            out[((size_t)b * NNODE + nm) * EMBED + col] = val;
        }
    }
}

// ---------------------------------------------------------------------------
extern "C" void kernel_launch(void* const* d_in, const int* in_sizes, int n_in,
                              void* d_out, int out_size, void* d_ws, size_t ws_size,
                              hipStream_t stream) {
    const float* locs = (const float*)d_in[0];   // [B][N][2]
    const int*   ei   = (const int*)  d_in[1];   // [B][2][E]
    const float* ea   = (const float*)d_in[2];   // [B][E][2]
    const float* W    = (const float*)d_in[3];   // [128][6]
    const float* bias = (const float*)d_in[4];   // [128]
    float*       out  = (float*)d_out;           // [B][N][128]
    float*       sums = (float*)d_ws;            // [B][N][8] = 2 MB

    const int nsum = BATCH * NNODE * 8;
    zero_ws_kernel<<<(nsum + 255) / 256, 256, 0, stream>>>(sums, nsum);

    dim3 gscat(NBLK_E, BATCH);
    edge_scatter_kernel<<<gscat, 256, 0, stream>>>(locs, ei, ea, sums);

    node_gemm_wmma_kernel<<<(BATCH * GROUPS) / 4, 128, 0, stream>>>(locs, sums, W, bias, out);
}